// CvxMPC_10660108829140
// MI455X (gfx1250) — compile-verified
//
#include <hip/hip_runtime.h>
#include <hip/hip_bf16.h>

typedef __attribute__((ext_vector_type(2))) float v2f;
typedef __attribute__((ext_vector_type(8))) float v8f;

#define N_DIM 512
#define M_DIM 128
#define HORIZON 50
#define Q_COST 0.01f
#define R_COST 0.01f
#define BATCH 32768

// ---------------------------------------------------------------------------
// Generic f32 WMMA GEMM:  D = alpha * op(A) @ B [+ Cin] [+ diag*I]
// One wave computes a 16x64 strip of D (4 accumulator tiles) using
// V_WMMA_F32_16X16X4_F32: each A fragment is reused across 4 B tiles
// (2.5 loads/WMMA) and the 4 chains are independent (XDL pipelining).
// Fragment layouts per CDNA5 ISA 7.12.2:
//   A (16x4): lanes 0-15 hold M=0..15; half-wave selects K pair (0,1)/(2,3)
//   B (4x16): lanes 0-15 hold N=0..15; half-wave selects K pair
//   C/D     : 8 VGPRs, M = vgpr + 8*(lane>=16), N = lane&15
// ---------------------------------------------------------------------------
template <bool TA, bool ADDC>
__global__ __launch_bounds__(256) void gemm_wmma_f32(
    const float* __restrict__ Am, int lda,
    const float* __restrict__ Bm, int ldb,
    const float* __restrict__ Cin,
    float* __restrict__ D, int ldd,
    int Mdim, int Ndim, int Kdim, float alpha, float diag) {
  int wave = (int)((blockIdx.x * blockDim.x + threadIdx.x) >> 5);
  int ntn = Ndim >> 6;  // 64-wide column groups
  int total = (Mdim >> 4) * ntn;
  if (wave >= total) return;  // whole-wave uniform: EXEC stays all-ones

  int tm = (wave / ntn) << 4;
  int tn = (wave % ntn) << 6;
  int lane = threadIdx.x & 31;
  int half = lane >> 4;  // selects K pair within a 4-wide K step
  int l15 = lane & 15;

  int arow = tm + l15;  // A-fragment M index for this lane

  v8f acc[4];
#pragma unroll
  for (int t = 0; t < 4; ++t) acc[t] = (v8f){};

  for (int k = 0; k < Kdim; k += 4) {
    int ka = k + half * 2;
    v2f a;
    if (TA) {  // column read, lane-coalesced (A[k][tm..tm+15] contiguous)
      a.x = Am[(size_t)(ka + 0) * lda + arow];
      a.y = Am[(size_t)(ka + 1) * lda + arow];
    } else {  // contiguous k-pair in one b64 load
      a = *(const v2f*)(Am + (size_t)arow * lda + ka);
    }
#pragma unroll
    for (int t = 0; t < 4; ++t) {
      int bcol = tn + t * 16 + l15;
      v2f b;
      b.x = Bm[(size_t)(ka + 0) * ldb + bcol];
      b.y = Bm[(size_t)(ka + 1) * ldb + bcol];
      acc[t] = __builtin_amdgcn_wmma_f32_16x16x4_f32(
          false, a, false, b, (short)0, acc[t], false, false);
    }
  }

  int orow = tm + half * 8;
#pragma unroll
  for (int t = 0; t < 4; ++t) {
    int ocol = tn + t * 16 + l15;
#pragma unroll
    for (int v = 0; v < 8; ++v) {
      float r = alpha * acc[t][v];
      if (ADDC) r += Cin[(size_t)(orow + v) * ldd + ocol];
      if ((orow + v) == ocol) r += diag;
      D[(size_t)(orow + v) * ldd + ocol] = r;
    }
  }
}

// ---------------------------------------------------------------------------
// Fused: in-place Cholesky of S (128x128 SPD), then solve (L L^T) K = G
// in-place in G (128x512). Single block of 512 threads:
//   phase 1: thread i<128 owns row i of the factorization (uniform barriers)
//   phase 2: thread c owns column c of the 512-column two-sided solve
// S reads in phase 2 are wave-uniform (broadcast); G accesses coalesced.
// ---------------------------------------------------------------------------
__global__ __launch_bounds__(512) void chol_solve(float* __restrict__ S,
                                                  float* __restrict__ G) {
  int tid = threadIdx.x;  // 512 threads
  for (int j = 0; j < 128; ++j) {
    if (tid == j) S[j * 128 + j] = sqrtf(S[j * 128 + j]);
    __syncthreads();
    float djj = S[j * 128 + j];
    if (tid > j && tid < 128) S[tid * 128 + j] /= djj;
    __syncthreads();
    if (tid > j && tid < 128) {
      float lij = S[tid * 128 + j];
      for (int kc = j + 1; kc <= tid; ++kc)
        S[tid * 128 + kc] -= lij * S[kc * 128 + j];
    }
    __syncthreads();
  }

  int c = tid;
  for (int i = 0; i < 128; ++i) {  // forward: L y = g
    float s = G[i * N_DIM + c];
    for (int j = 0; j < i; ++j) s -= S[i * 128 + j] * G[j * N_DIM + c];
    G[i * N_DIM + c] = s / S[i * 128 + i];
  }
  for (int i = 127; i >= 0; --i) {  // backward: L^T k = y
    float s = G[i * N_DIM + c];
    for (int j = i + 1; j < 128; ++j) s -= S[j * 128 + i] * G[j * N_DIM + c];
    G[i * N_DIM + c] = s / S[i * 128 + i];
  }
}

__global__ void init_P(float* __restrict__ P) {
  int idx = (int)(blockIdx.x * blockDim.x + threadIdx.x);
  int r = idx >> 9, c = idx & 511;
  P[idx] = (r == c) ? Q_COST : 0.0f;
}

// ---------------------------------------------------------------------------
// out[32768,128] = -obs[32768,512] @ K0[128,512]^T
// Block = 256 threads = 8 waves; block covers 128 obs rows x all 128 cols.
// K0 K-chunks (128 rows x 64 K) staged in LDS (pitch-padded), obs streamed
// exactly once from HBM. 8 independent WMMA chains per wave.
// ---------------------------------------------------------------------------
#define KC 64
#define BPITCH 66
__global__ __launch_bounds__(256) void final_gemm(
    const float* __restrict__ obs, const float* __restrict__ K0,
    float* __restrict__ out) {
  __shared__ float ldsB[128 * BPITCH];  // ~33 KB

  int tid = threadIdx.x;
  int waveId = tid >> 5;
  int lane = tid & 31;
  int half = lane >> 4;
  int l15 = lane & 15;
  int rowBase = (int)blockIdx.x * 128 + waveId * 16;

  const float* obsRow = obs + (size_t)(rowBase + l15) * N_DIM;

  v8f acc[8];
#pragma unroll
  for (int t = 0; t < 8; ++t) acc[t] = (v8f){};

  for (int k0 = 0; k0 < N_DIM; k0 += KC) {
    __syncthreads();
    // stage K0[:, k0:k0+KC] -> LDS as [n][kk], float2-wide, coalesced
    for (int idx = tid; idx < 128 * (KC / 2); idx += 256) {
      int n = idx / (KC / 2);
      int k2 = idx % (KC / 2);
      float2 d = *(const float2*)(K0 + (size_t)n * N_DIM + k0 + k2 * 2);
      *(float2*)(&ldsB[n * BPITCH + k2 * 2]) = d;
    }
    __syncthreads();

    if (k0 + KC < N_DIM)  // prefetch next obs chunk for this lane's row
      __builtin_prefetch(obsRow + k0 + KC, 0, 0);

    for (int kk = 0; kk < KC; kk += 4) {
      int ka = kk + half * 2;
      v2f a = *(const v2f*)(obsRow + k0 + ka);
#pragma unroll
      for (int t = 0; t < 8; ++t) {
        int n = t * 16 + l15;
        v2f b = *(const v2f*)(&ldsB[n * BPITCH + ka]);
        acc[t] = __builtin_amdgcn_wmma_f32_16x16x4_f32(
            false, a, false, b, (short)0, acc[t], false, false);
      }
    }
  }

#pragma unroll
  for (int t = 0; t < 8; ++t) {
#pragma unroll
    for (int v = 0; v < 8; ++v) {
      out[(size_t)(rowBase + half * 8 + v) * M_DIM + t * 16 + l15] = -acc[t][v];
    }
  }
}

// ---------------------------------------------------------------------------
// Host orchestration
// ---------------------------------------------------------------------------
template <bool TA, bool ADDC>
static void launch_gemm(const float* Am, int lda, const float* Bm, int ldb,
                        const float* Cin, float* D, int ldd, int Mdim, int Ndim,
                        int Kdim, float alpha, float diag, hipStream_t stream) {
  int waves = (Mdim / 16) * (Ndim / 64);
  int blocks = (waves + 7) / 8;  // 8 waves per 256-thread block
  gemm_wmma_f32<TA, ADDC><<<blocks, 256, 0, stream>>>(
      Am, lda, Bm, ldb, Cin, D, ldd, Mdim, Ndim, Kdim, alpha, diag);
}

extern "C" void kernel_launch(void* const* d_in, const int* in_sizes, int n_in,
                              void* d_out, int out_size, void* d_ws,
                              size_t ws_size, hipStream_t stream) {
  const float* obs = (const float*)d_in[0];  // [32768,512]
  const float* A = (const float*)d_in[1];    // [512,512]
  const float* B = (const float*)d_in[2];    // [512,128]
  float* out = (float*)d_out;                // [32768,128]

  float* ws = (float*)d_ws;
  float* P0 = ws;                    // 512*512
  float* P1 = P0 + N_DIM * N_DIM;    // 512*512
  float* BtP = P1 + N_DIM * N_DIM;   // 128*512
  float* S = BtP + M_DIM * N_DIM;    // 128*128
  float* G = S + M_DIM * M_DIM;      // 128*512 (becomes K in-place)
  float* Mx = G + M_DIM * N_DIM;     // 512*512
  float* T = Mx + N_DIM * N_DIM;     // 512*512

  init_P<<<(N_DIM * N_DIM) / 256, 256, 0, stream>>>(P0);

  const int steps = HORIZON - 1;  // 49
  for (int step = 0; step < steps; ++step) {
    float* Pc = (step & 1) ? P1 : P0;
    float* Pn = (step & 1) ? P0 : P1;

    // BtP = B^T @ P            [128,512]
    launch_gemm<true, false>(B, M_DIM, Pc, N_DIM, nullptr, BtP, N_DIM, M_DIM,
                             N_DIM, N_DIM, 1.0f, 0.0f, stream);
    // S = R + BtP @ B          [128,128]
    launch_gemm<false, false>(BtP, N_DIM, B, M_DIM, nullptr, S, M_DIM, M_DIM,
                              M_DIM, N_DIM, 1.0f, R_COST, stream);
    // G = BtP @ A              [128,512]
    launch_gemm<false, false>(BtP, N_DIM, A, N_DIM, nullptr, G, N_DIM, M_DIM,
                              N_DIM, N_DIM, 1.0f, 0.0f, stream);
    // chol(S); G <- K = (R + B'PB)^-1 G
    chol_solve<<<1, 512, 0, stream>>>(S, G);

    if (step == steps - 1) break;  // last K is K0; P update unused

    // Mx = A - B @ K           [512,512]
    launch_gemm<false, true>(B, M_DIM, G, N_DIM, A, Mx, N_DIM, N_DIM, N_DIM,
                             M_DIM, -1.0f, 0.0f, stream);
    // T = P @ Mx               [512,512]
    launch_gemm<false, false>(Pc, N_DIM, Mx, N_DIM, nullptr, T, N_DIM, N_DIM,
                              N_DIM, N_DIM, 1.0f, 0.0f, stream);
    // Pn = Q + A^T @ T         [512,512]
    launch_gemm<true, false>(A, N_DIM, T, N_DIM, nullptr, Pn, N_DIM, N_DIM,
                             N_DIM, N_DIM, 1.0f, Q_COST, stream);
  }

  // u0 = -obs @ K0^T
  final_gemm<<<BATCH / 128, 256, 0, stream>>>(obs, G, out);
}